// SparseGNNLayer_64209761075733
// MI455X (gfx1250) — compile-verified
//
#include <hip/hip_runtime.h>

#define F 32          // F_IN == F_OUT == 32 == wave32 width
#define CHUNK 128     // edges per wave in the SpMM phase

typedef float v2f __attribute__((ext_vector_type(2)));
typedef float v8f __attribute__((ext_vector_type(8)));

// ---------------------------------------------------------------------------
// Phase 0: zero the aggregation buffer (d_out is reused as agg storage).
// ---------------------------------------------------------------------------
__global__ __launch_bounds__(256) void gnn_zero_kernel(float4* __restrict__ out, int n4) {
    int i = blockIdx.x * blockDim.x + threadIdx.x;
    if (i < n4) out[i] = make_float4(0.f, 0.f, 0.f, 0.f);
}

// ---------------------------------------------------------------------------
// Phase 1: COO SpMM scatter-add.  One wave per CHUNK edges; lane = feature.
// edge_rows is sorted, so we accumulate in registers while the destination
// row is unchanged and flush with a native f32 global atomic on transitions.
// ---------------------------------------------------------------------------
__global__ __launch_bounds__(256) void gnn_spmm_kernel(
        const float* __restrict__ x,
        const int*   __restrict__ rows,
        const int*   __restrict__ cols,
        const float* __restrict__ vals,
        float*       __restrict__ agg,
        int n_edges) {
    const int lane = threadIdx.x & 31;
    const int wave = (blockIdx.x * blockDim.x + threadIdx.x) >> 5;
    const int start = wave * CHUNK;
    if (start >= n_edges) return;
    const int end = min(start + CHUNK, n_edges);

    int   cur_row = __builtin_amdgcn_readfirstlane(rows[start]);
    float acc = 0.f;

    for (int e = start; e < end; ++e) {
        const int   r = __builtin_amdgcn_readfirstlane(rows[e]);
        const int   c = __builtin_amdgcn_readfirstlane(cols[e]);
        const float v = vals[e];
        if (r != cur_row) {
            unsafeAtomicAdd(&agg[(long long)cur_row * F + lane], acc);
            acc = 0.f;
            cur_row = r;
        }
        acc += x[(long long)c * F + lane] * v;
    }
    unsafeAtomicAdd(&agg[(long long)cur_row * F + lane], acc);
}

// ---------------------------------------------------------------------------
// Phase 2: out = relu(agg @ W^T + b), in place over `io` (agg in, out out).
// One wave per 16-row tile.  K=32 chained as 8x V_WMMA_F32_16X16X4_F32 into
// two 16x16 f32 accumulators (output cols 0-15 and 16-31).
//
// ISA operand layouts (wave32):
//   A 16x4 f32 : lane<16 -> M=lane,  v.x=K0, v.y=K1 ; lane>=16 -> K2,K3
//   B 4x16 f32 : lane<16 -> N=lane,  v.x=K0, v.y=K1 ; lane>=16 -> K2,K3
//   C/D 16x16  : VGPR g, lane l -> M = g + 8*(l/16), N = l%16
// B[k][n] = W[n][k]  (since out = agg @ W^T, W row-major [out][in]).
// ---------------------------------------------------------------------------
__global__ __launch_bounds__(256) void gnn_gemm_wmma_kernel(
        float*       __restrict__ io,
        const float* __restrict__ W,
        const float* __restrict__ bias,
        int n_tiles) {
    const int lane = threadIdx.x & 31;
    const int tile = blockIdx.x * 8 + (threadIdx.x >> 5);
    if (tile >= n_tiles) return;

    const int m0   = tile * 16;
    const int half = lane >> 4;   // 0: K pair {0,1}, 1: K pair {2,3}
    const int lm   = lane & 15;   // M for A loads, N for B loads

    v8f c0 = {};
    v8f c1 = {};
    const float* arow = io + (long long)(m0 + lm) * F;

#pragma unroll
    for (int kb = 0; kb < 8; ++kb) {
        const int k0 = kb * 4 + half * 2;
        v2f a, b0, b1;
        a.x  = arow[k0];
        a.y  = arow[k0 + 1];
        b0.x = W[lm * F + k0];
        b0.y = W[lm * F + k0 + 1];
        b1.x = W[(16 + lm) * F + k0];
        b1.y = W[(16 + lm) * F + k0 + 1];
        c0 = __builtin_amdgcn_wmma_f32_16x16x4_f32(
                 false, a, false, b0, (short)0, c0, false, false);
        c1 = __builtin_amdgcn_wmma_f32_16x16x4_f32(
                 false, a, false, b1, (short)0, c1, false, false);
    }

    const float bias0 = bias[lm];
    const float bias1 = bias[16 + lm];

#pragma unroll
    for (int g = 0; g < 8; ++g) {
        const int row = m0 + g + 8 * half;
        float o0 = c0[g] + bias0;  o0 = o0 > 0.f ? o0 : 0.f;
        float o1 = c1[g] + bias1;  o1 = o1 > 0.f ? o1 : 0.f;
        io[(long long)row * F + lm]      = o0;
        io[(long long)row * F + 16 + lm] = o1;
    }
}

// Scalar tail for rows not covered by full 16-row tiles (N%16 != 0).
// With N_NODES = 100000 this launches zero blocks.
__global__ __launch_bounds__(64) void gnn_gemm_tail_kernel(
        float* __restrict__ io,
        const float* __restrict__ W,
        const float* __restrict__ bias,
        int row_start, int n_rows) {
    const int row  = row_start + blockIdx.x;
    const int lane = threadIdx.x & 31;
    if (row >= row_start + n_rows || threadIdx.x >= 32) return;
    float a[F];
#pragma unroll
    for (int k = 0; k < F; ++k) a[k] = io[(long long)row * F + k];
    float s = bias[lane];
#pragma unroll
    for (int k = 0; k < F; ++k) s += a[k] * W[lane * F + k];
    io[(long long)row * F + lane] = s > 0.f ? s : 0.f;
}

// ---------------------------------------------------------------------------
extern "C" void kernel_launch(void* const* d_in, const int* in_sizes, int n_in,
                              void* d_out, int out_size, void* d_ws, size_t ws_size,
                              hipStream_t stream) {
    const float* x    = (const float*)d_in[0];
    const int*   rows = (const int*)  d_in[1];
    const int*   cols = (const int*)  d_in[2];
    const float* vals = (const float*)d_in[3];
    const float* W    = (const float*)d_in[4];
    const float* bias = (const float*)d_in[5];
    float* out = (float*)d_out;

    const int n_nodes = in_sizes[0] / F;
    const int n_edges = in_sizes[1];

    // Phase 0: zero agg (stored in d_out).
    {
        const int n4 = (n_nodes * F) / 4;
        const int blocks = (n4 + 255) / 256;
        gnn_zero_kernel<<<blocks, 256, 0, stream>>>((float4*)out, n4);
    }

    // Phase 1: SpMM scatter-add into agg.
    {
        const int n_waves = (n_edges + CHUNK - 1) / CHUNK;
        const int blocks  = (n_waves + 7) / 8;   // 8 waves per 256-thread block
        gnn_spmm_kernel<<<blocks, 256, 0, stream>>>(x, rows, cols, vals, out, n_edges);
    }

    // Phase 2: WMMA linear + bias + relu, in place.
    {
        const int n_tiles = n_nodes / 16;
        if (n_tiles > 0) {
            const int blocks = (n_tiles + 7) / 8;
            gnn_gemm_wmma_kernel<<<blocks, 256, 0, stream>>>(out, W, bias, n_tiles);
        }
        const int rem = n_nodes - n_tiles * 16;
        if (rem > 0) {
            gnn_gemm_tail_kernel<<<rem, 64, 0, stream>>>(out, W, bias, n_tiles * 16, rem);
        }
    }
}